// NestedGIN_87393994539011
// MI455X (gfx1250) — compile-verified
//
#include <hip/hip_runtime.h>
#include <hip/hip_bf16.h>

#define N_NODES   500000
#define E_EDGES   4000000
#define H         64
#define OUT_DIM   16
#define NUM_SUB   10000
#define NUM_GRAPH 100
#define BN_EPS    1e-5f

typedef float v2f __attribute__((ext_vector_type(2)));
typedef float v8f __attribute__((ext_vector_type(8)));

// ---------------------------------------------------------------- utilities
__global__ void fill0_kernel(float* __restrict__ p, long long n) {
  long long i = (long long)blockIdx.x * blockDim.x + threadIdx.x;
  long long stride = (long long)gridDim.x * blockDim.x;
  for (; i < n; i += stride) p[i] = 0.0f;
}

// ------------------------------------------------- edge aggregation (layer 1, scalar feature)
__global__ void edge_agg_scalar(const float* __restrict__ z,
                                const int* __restrict__ src,
                                const int* __restrict__ dst,
                                float* __restrict__ agg, int e) {
  int i = blockIdx.x * blockDim.x + threadIdx.x;
  if (i < e) atomicAdd(&agg[dst[i]], z[src[i]]);
}

// ------------------------------------------------- edge aggregation, H=64 features per node
__global__ void edge_agg_h(const float* __restrict__ x,
                           const int* __restrict__ src,
                           const int* __restrict__ dst,
                           float* __restrict__ agg, int e) {
  int i = blockIdx.x * blockDim.x + threadIdx.x;
  if (i >= e) return;
  const float4* xs = (const float4*)(x + (long long)src[i] * H);
  float* ad = agg + (long long)dst[i] * H;
#pragma unroll
  for (int k = 0; k < H / 4; ++k) {
    float4 v = xs[k];
    atomicAdd(&ad[k * 4 + 0], v.x);
    atomicAdd(&ad[k * 4 + 1], v.y);
    atomicAdd(&ad[k * 4 + 2], v.z);
    atomicAdd(&ad[k * 4 + 3], v.w);
  }
}

// ------------------------------------------------- layer-1 first MLP stage: h = relu((z+agg) * W1a + b1a)
__global__ void layer1_pre(const float* __restrict__ z, const float* __restrict__ aggz,
                           const float* __restrict__ W1a, const float* __restrict__ b1a,
                           float* __restrict__ h, long long total) {
  long long i = (long long)blockIdx.x * blockDim.x + threadIdx.x;
  if (i >= total) return;
  int node = (int)(i >> 6);
  int j = (int)(i & 63);
  float s = z[node] + aggz[node];
  float v = s * W1a[j] + b1a[j];
  h[i] = v > 0.0f ? v : 0.0f;
}

// ------------------------------------------------- fp32 WMMA GEMM:  Out = relu((A [+ Agg]) @ W + bias)
// A: [nRows, 64], W: [64, 64] row-major (k,n), Out: [nRows, 64]
// One wave computes a 16-row strip across all 64 output columns (4 WMMA tiles).
// Block = 8 waves = 128 rows. K-loop: 16 steps x 4 V_WMMA_F32_16X16X4_F32.
// Weights staged in LDS pre-paired as float2 {W[2k][n], W[2k+1][n]} so each B
// fragment is one aligned ds_load_b64 directly into the WMMA operand pair.
template <int USE_AGG>
__global__ __launch_bounds__(256)
void gemm_relu_wmma(const float* __restrict__ A,
                    const float* __restrict__ Agg,
                    const float* __restrict__ W,
                    const float* __restrict__ bias,
                    float* __restrict__ Out,
                    int nRows) {
  __shared__ v2f   sWp[(H / 2) * H];   // [kpair][n] -> {W[2kp][n], W[2kp+1][n]}
  __shared__ float sB[H];
  for (int idx = threadIdx.x; idx < (H / 2) * H; idx += 256) {
    int kp = idx >> 6;        // 0..31
    int n  = idx & 63;
    v2f w;
    w.x = W[(2 * kp) * H + n];
    w.y = W[(2 * kp + 1) * H + n];
    sWp[idx] = w;
  }
  if (threadIdx.x < H) sB[threadIdx.x] = bias[threadIdx.x];
  __syncthreads();

  const int lane    = threadIdx.x & 31;
  const int wave    = threadIdx.x >> 5;
  const int rowBase = blockIdx.x * 128 + wave * 16;
  const int halfSel = lane >> 4;   // selects K pair within the 16x4 A fragment
  const int lidx    = lane & 15;

  v8f c0 = {}; v8f c1 = {}; v8f c2 = {}; v8f c3 = {};

  // A fragment row for this lane (clamped: WMMA needs EXEC all-ones; stores are masked later)
  int arow = rowBase + lidx;
  if (arow >= nRows) arow = nRows - 1;
  const float* arp = A + (long long)arow * H;
  const float* agp = Agg + (long long)arow * H;

#pragma unroll
  for (int kk = 0; kk < 16; ++kk) {
    const int kpair = kk * 2 + halfSel;          // K pair index 0..31
    const int kb    = kpair * 2;                 // even K base

    v2f a = *(const v2f*)(arp + kb);
    if (USE_AGG) {
      v2f g = *(const v2f*)(agp + kb);
      a.x += g.x; a.y += g.y;
    }

    const v2f* brow = &sWp[kpair * H + lidx];
    v2f b0 = brow[0];
    v2f b1 = brow[16];
    v2f b2 = brow[32];
    v2f b3 = brow[48];

    c0 = __builtin_amdgcn_wmma_f32_16x16x4_f32(false, a, false, b0, (short)0, c0, false, false);
    c1 = __builtin_amdgcn_wmma_f32_16x16x4_f32(false, a, false, b1, (short)0, c1, false, false);
    c2 = __builtin_amdgcn_wmma_f32_16x16x4_f32(false, a, false, b2, (short)0, c2, false, false);
    c3 = __builtin_amdgcn_wmma_f32_16x16x4_f32(false, a, false, b3, (short)0, c3, false, false);
  }

  // epilogue: bias + relu + masked store.  C layout: VGPR r, lane L -> M = r + 8*(L>>4), N = L&15
#pragma unroll
  for (int t = 0; t < 4; ++t) {
    v8f c = (t == 0) ? c0 : (t == 1) ? c1 : (t == 2) ? c2 : c3;
    int n = t * 16 + lidx;
    float bv = sB[n];
#pragma unroll
    for (int r = 0; r < 8; ++r) {
      int row = rowBase + r + halfSel * 8;
      if (row < nRows) {
        float v = c[r] + bv;
        Out[(long long)row * H + n] = v > 0.0f ? v : 0.0f;
      }
    }
  }
}

// ------------------------------------------------- segment-sum pooling (node->sub, sub->graph)
__global__ void pool_kernel(const float* __restrict__ x, const int* __restrict__ map,
                            float* __restrict__ outp, long long total) {
  long long i = (long long)blockIdx.x * blockDim.x + threadIdx.x;
  if (i >= total) return;
  int rowi = (int)(i >> 6);
  int j = (int)(i & 63);
  atomicAdd(&outp[(long long)map[rowi] * H + j], x[i]);
}

// ------------------------------------------------- head: relu(G@Wl1+bl1) @ Wl2 + bl2, then batchnorm
__global__ __launch_bounds__(256)
void head_kernel(const float* __restrict__ G,
                 const float* __restrict__ Wl1, const float* __restrict__ bl1,
                 const float* __restrict__ Wl2, const float* __restrict__ bl2,
                 float* __restrict__ out) {
  __shared__ float sg[NUM_GRAPH * H];
  __shared__ float sy[NUM_GRAPH * OUT_DIM];
  __shared__ float smean[OUT_DIM];
  __shared__ float srstd[OUT_DIM];

  for (int idx = threadIdx.x; idx < NUM_GRAPH * H; idx += blockDim.x) {
    int r = idx / H, c = idx % H;
    float acc = bl1[c];
    for (int k = 0; k < H; ++k) acc += G[r * H + k] * Wl1[k * H + c];
    sg[idx] = acc > 0.0f ? acc : 0.0f;
  }
  __syncthreads();

  for (int idx = threadIdx.x; idx < NUM_GRAPH * OUT_DIM; idx += blockDim.x) {
    int r = idx / OUT_DIM, c = idx % OUT_DIM;
    float acc = bl2[c];
    for (int k = 0; k < H; ++k) acc += sg[r * H + k] * Wl2[k * OUT_DIM + c];
    sy[idx] = acc;
  }
  __syncthreads();

  if (threadIdx.x < OUT_DIM) {
    int c = threadIdx.x;
    float m = 0.0f;
    for (int r = 0; r < NUM_GRAPH; ++r) m += sy[r * OUT_DIM + c];
    m /= (float)NUM_GRAPH;
    float s = 0.0f;
    for (int r = 0; r < NUM_GRAPH; ++r) { float d = sy[r * OUT_DIM + c] - m; s += d * d; }
    s /= (float)NUM_GRAPH;
    smean[c] = m;
    srstd[c] = rsqrtf(s + BN_EPS);
  }
  __syncthreads();

  for (int idx = threadIdx.x; idx < NUM_GRAPH * OUT_DIM; idx += blockDim.x) {
    int c = idx % OUT_DIM;
    out[idx] = (sy[idx] - smean[c]) * srstd[c];
  }
}

// ---------------------------------------------------------------- launch
static inline int cdiv_i(long long a, long long b) { return (int)((a + b - 1) / b); }

extern "C" void kernel_launch(void* const* d_in, const int* in_sizes, int n_in,
                              void* d_out, int out_size, void* d_ws, size_t ws_size,
                              hipStream_t stream) {
  (void)in_sizes; (void)n_in; (void)out_size; (void)ws_size;

  const float* z   = (const float*)d_in[0];
  const int*   ei  = (const int*)d_in[1];
  const int*   src = ei;              // edge_index[0]
  const int*   dst = ei + E_EDGES;    // edge_index[1]
  const int*   n2s = (const int*)d_in[2];
  const int*   s2g = (const int*)d_in[3];
  const float* W1a = (const float*)d_in[4];
  const float* b1a = (const float*)d_in[5];
  const float* W1b = (const float*)d_in[6];
  const float* b1b = (const float*)d_in[7];
  const float* Wsa = (const float*)d_in[8];   // [3,64,64]
  const float* bsa = (const float*)d_in[9];   // [3,64]
  const float* Wsb = (const float*)d_in[10];  // [3,64,64]
  const float* bsb = (const float*)d_in[11];  // [3,64]
  const float* Wl1 = (const float*)d_in[12];
  const float* bl1 = (const float*)d_in[13];
  const float* Wl2 = (const float*)d_in[14];
  const float* bl2 = (const float*)d_in[15];
  float* out = (float*)d_out;

  // workspace layout (floats)
  float* ws       = (float*)d_ws;
  float* bufX     = ws;                                   // [N,64]
  float* bufAgg   = ws + (long long)N_NODES * H;          // [N,64]
  float* bufSub   = ws + 2LL * N_NODES * H;               // [NUM_SUB,64]  (also reused as aggz[N])
  float* bufGraph = bufSub + (long long)NUM_SUB * H;      // [100,64]

  const int TB = 256;
  const long long NH = (long long)N_NODES * H;
  const int gemmBlocks = (N_NODES + 127) / 128;

  // ---- layer 1 (input feature is scalar per node) ----
  fill0_kernel<<<cdiv_i(N_NODES, TB), TB, 0, stream>>>(bufSub, N_NODES);
  edge_agg_scalar<<<cdiv_i(E_EDGES, TB), TB, 0, stream>>>(z, src, dst, bufSub, E_EDGES);
  layer1_pre<<<cdiv_i(NH, TB), TB, 0, stream>>>(z, bufSub, W1a, b1a, bufAgg, NH);
  gemm_relu_wmma<0><<<gemmBlocks, 256, 0, stream>>>(bufAgg, nullptr, W1b, b1b, bufX, N_NODES);

  // ---- layers 2..4 ----
  for (int l = 0; l < 3; ++l) {
    fill0_kernel<<<cdiv_i(NH, TB), TB, 0, stream>>>(bufAgg, NH);
    edge_agg_h<<<cdiv_i(E_EDGES, TB), TB, 0, stream>>>(bufX, src, dst, bufAgg, E_EDGES);
    // h = relu((x + agg) @ Wsa + bsa), written in place over agg (each tile reads only its own rows first)
    gemm_relu_wmma<1><<<gemmBlocks, 256, 0, stream>>>(bufX, bufAgg, Wsa + l * H * H, bsa + l * H,
                                                      bufAgg, N_NODES);
    // x = relu(h @ Wsb + bsb)
    gemm_relu_wmma<0><<<gemmBlocks, 256, 0, stream>>>(bufAgg, nullptr, Wsb + l * H * H, bsb + l * H,
                                                      bufX, N_NODES);
  }

  // ---- pooling ----
  fill0_kernel<<<cdiv_i((long long)NUM_SUB * H, TB), TB, 0, stream>>>(bufSub, (long long)NUM_SUB * H);
  fill0_kernel<<<cdiv_i(NUM_GRAPH * H, TB), TB, 0, stream>>>(bufGraph, NUM_GRAPH * H);
  pool_kernel<<<cdiv_i(NH, TB), TB, 0, stream>>>(bufX, n2s, bufSub, NH);
  pool_kernel<<<cdiv_i((long long)NUM_SUB * H, TB), TB, 0, stream>>>(bufSub, s2g, bufGraph,
                                                                     (long long)NUM_SUB * H);

  // ---- head + batchnorm ----
  head_kernel<<<1, 256, 0, stream>>>(bufGraph, Wl1, bl1, Wl2, bl2, out);
}